// dwt_scam_38371237822845
// MI455X (gfx1250) — compile-verified
//
#include <hip/hip_runtime.h>
#include <math.h>

// ---------------------------------------------------------------------------
// Fused DWT-attention block for MI455X (gfx1250, wave32).
// Bandwidth-bound problem (~670 MB HBM traffic @ 23.3 TB/s ~= 29us floor);
// the 64x64 conv1x1 GEMM is done with V_WMMA_F32_16X16X4_F32 (exact fp32).
// All reductions are fixed-order (shuffle butterflies + ordered partial
// buffers) -> bitwise deterministic, no float atomics.
// Workspace requirement: ~142 MB (fp32).
// ---------------------------------------------------------------------------

typedef float v2f __attribute__((ext_vector_type(2)));
typedef float v8f __attribute__((ext_vector_type(8)));

#define Cc    64
#define Hx    256
#define Wxx   256
#define hA    128
#define wAq   128
#define HWA   16384     // 128*128 (A-space pixels)
#define XHW   65536     // 256*256 (x-space pixels)
#define NB    8

__device__ __forceinline__ float siluf(float y) { return y / (1.f + __expf(-y)); }
__device__ __forceinline__ float sigm(float y)  { return 1.f / (1.f + __expf(-y)); }

// ---------------------------------------------------------------------------
// Kernel 1: fused conv1x1(WMMA) + silu + dwt(v) + dwt(x) logits + A partials.
// Block: 256 threads (8 waves). Covers (b, A-row i, 32 A-cols) = 128 x-pixels,
// all 64 channels.
// ---------------------------------------------------------------------------
__global__ __launch_bounds__(256) void k1_conv_dwt(
    const float* __restrict__ x,
    const float* __restrict__ conv_w, const float* __restrict__ conv_s, const float* __restrict__ conv_t,
    const float* __restrict__ cAw, const float* __restrict__ cAs, const float* __restrict__ cAt,
    const float* __restrict__ cBw, const float* __restrict__ cBs, const float* __restrict__ cBt,
    const float* __restrict__ cCw, const float* __restrict__ cCs, const float* __restrict__ cCt,
    const float* __restrict__ cDw, const float* __restrict__ cDs, const float* __restrict__ cDt,
    float* __restrict__ A1, float* __restrict__ B1, float* __restrict__ C1, float* __restrict__ D1,
    float* __restrict__ lA, float* __restrict__ lB, float* __restrict__ lC, float* __restrict__ lD,
    float* __restrict__ sumPart, float* __restrict__ maxPart)
{
    __shared__ float wLds[64][66];      // conv_w, padded vs bank conflicts
    __shared__ float vLds[64][128];     // silu(conv) result, [ch][pixel]
    __shared__ float dotW[8][4][32];    // per-wave logit partials

    const int tid  = threadIdx.x;
    const int lane = tid & 31;
    const int wv   = tid >> 5;
    const int jt   = blockIdx.x;        // 0..3  (32 A-cols each)
    const int i    = blockIdx.y;        // 0..127 (A-row)
    const int b    = blockIdx.z;        // 0..7
    const int c0   = jt * 64;           // x-col base

    for (int idx = tid; idx < 64 * 64; idx += 256)
        wLds[idx >> 6][idx & 63] = conv_w[idx];
    __syncthreads();

    // ---- WMMA GEMM: v[ch,p] = sum_k conv_w[ch,k] * x[k,p], 128 pixels ----
    // pixel p: p<64 -> row 2i, col c0+p ; p>=64 -> row 2i+1, col c0+p-64
    const int p     = wv * 16 + (lane & 15);
    const int xrow  = 2 * i + (p >> 6);
    const int xcol  = c0 + (p & 63);
    const int khalf = (lane >> 4) * 2;  // K sub-offset per ISA f32 frag layout
    const float* xb = x + (size_t)b * Cc * XHW + (size_t)xrow * Wxx + xcol;

    v8f acc[4];
    #pragma unroll
    for (int mt = 0; mt < 4; ++mt)
        #pragma unroll
        for (int r = 0; r < 8; ++r) acc[mt][r] = 0.f;

    #pragma unroll 4
    for (int k = 0; k < 16; ++k) {
        const int kc = 4 * k + khalf;
        v2f bf = { xb[(size_t)kc * XHW], xb[(size_t)(kc + 1) * XHW] };
        #pragma unroll
        for (int mt = 0; mt < 4; ++mt) {
            const int m = mt * 16 + (lane & 15);
            v2f af = { wLds[m][kc], wLds[m][kc + 1] };
            acc[mt] = __builtin_amdgcn_wmma_f32_16x16x4_f32(
                false, af, false, bf, (short)0, acc[mt], false, false);
        }
    }

    // epilogue: scale/bias + silu, into LDS (C/D layout: M = r + 8*(lane>=16))
    const int half8 = (lane >> 4) * 8;
    #pragma unroll
    for (int mt = 0; mt < 4; ++mt)
        #pragma unroll
        for (int r = 0; r < 8; ++r) {
            const int ch = mt * 16 + half8 + r;
            vLds[ch][p] = siluf(acc[mt][r] * conv_s[ch] + conv_t[ch]);
        }
    __syncthreads();

    // ---- dwt(v) -> A1..D1, dwt(x) -> logits + A sum/max partials ----
    const int kb = i * 4 + jt;                 // block index within batch, 0..511
    const int aj = lane;                       // A-col within tile
    const size_t pixBase = (size_t)i * wAq + (size_t)jt * 32 + aj;
    float rd0 = 0.f, rd1 = 0.f, rd2 = 0.f, rd3 = 0.f;

    for (int it = 0; it < 8; ++it) {
        const int ch = it * 8 + wv;
        const float v00 = vLds[ch][2 * aj],      v01 = vLds[ch][2 * aj + 1];
        const float v10 = vLds[ch][64 + 2 * aj], v11 = vLds[ch][64 + 2 * aj + 1];
        const float a1v = 0.5f * ( v00 + v10 + v01 + v11);
        const float b1v = 0.5f * (-v00 - v10 + v01 + v11);
        const float c1v = 0.5f * (-v00 + v10 - v01 + v11);
        const float d1v = 0.5f * ( v00 - v10 - v01 + v11);
        const size_t oo = ((size_t)b * Cc + ch) * HWA + pixBase;
        A1[oo] = a1v; B1[oo] = b1v; C1[oo] = c1v; D1[oo] = d1v;

        const float* xc = x + ((size_t)b * Cc + ch) * XHW + (size_t)(2 * i) * Wxx + (c0 + 2 * aj);
        const float x00 = xc[0], x01 = xc[1], x10 = xc[Wxx], x11 = xc[Wxx + 1];
        const float Ax = 0.5f * ( x00 + x10 + x01 + x11);
        const float Bx = 0.5f * (-x00 - x10 + x01 + x11);
        const float Cx = 0.5f * (-x00 + x10 - x01 + x11);
        const float Dx = 0.5f * ( x00 - x10 - x01 + x11);

        rd0 += cAw[ch] * Ax;
        rd1 += cBw[ch] * 0.5f * (Bx + Ax);
        rd2 += cCw[ch] * 0.5f * (Cx + Ax);
        rd3 += cDw[ch] * 0.5f * (Dx + Ax);

        float s = Ax, mm = Ax;
        #pragma unroll
        for (int off = 16; off; off >>= 1) {
            s  += __shfl_xor(s, off, 32);
            mm  = fmaxf(mm, __shfl_xor(mm, off, 32));
        }
        if (lane == 0) {
            sumPart[((size_t)b * 512 + kb) * 64 + ch] = s;
            maxPart[((size_t)b * 512 + kb) * 64 + ch] = mm;
        }
    }
    dotW[wv][0][lane] = rd0; dotW[wv][1][lane] = rd1;
    dotW[wv][2][lane] = rd2; dotW[wv][3][lane] = rd3;
    __syncthreads();

    if (tid < 128) {
        const int T = tid >> 5, a2 = tid & 31;
        float d = 0.f;
        #pragma unroll
        for (int w8 = 0; w8 < 8; ++w8) d += dotW[w8][T][a2];
        const float sT = (T == 0) ? cAs[0] : (T == 1) ? cBs[0] : (T == 2) ? cCs[0] : cDs[0];
        const float tT = (T == 0) ? cAt[0] : (T == 1) ? cBt[0] : (T == 2) ? cCt[0] : cDt[0];
        float* lp = (T == 0) ? lA : (T == 1) ? lB : (T == 2) ? lC : lD;
        lp[(size_t)b * HWA + (size_t)i * wAq + jt * 32 + a2] = siluf(d * sT + tT);
    }
}

// ---------------------------------------------------------------------------
// Kernel 2: finalize A mean/max over space; channel softmax -> A_gap, A_gmp.
// ---------------------------------------------------------------------------
__global__ __launch_bounds__(64) void k2_gap(
    const float* __restrict__ sumPart, const float* __restrict__ maxPart,
    float* __restrict__ gapW, float* __restrict__ gmpW)
{
    const int b = blockIdx.x, ch = threadIdx.x;
    float s = 0.f, m = -3.0e38f;
    for (int k = 0; k < 512; ++k) {
        s += sumPart[((size_t)b * 512 + k) * 64 + ch];
        m  = fmaxf(m, maxPart[((size_t)b * 512 + k) * 64 + ch]);
    }
    const float mean = s * (1.f / 16384.f);
    __shared__ float red[64];

    red[ch] = mean; __syncthreads();
    for (int off = 32; off; off >>= 1) { if (ch < off) red[ch] = fmaxf(red[ch], red[ch + off]); __syncthreads(); }
    const float mx1 = red[0]; __syncthreads();
    const float e1 = __expf(mean - mx1);
    red[ch] = e1; __syncthreads();
    for (int off = 32; off; off >>= 1) { if (ch < off) red[ch] += red[ch + off]; __syncthreads(); }
    const float sm1 = red[0]; __syncthreads();
    gapW[b * 64 + ch] = e1 / sm1;

    red[ch] = m; __syncthreads();
    for (int off = 32; off; off >>= 1) { if (ch < off) red[ch] = fmaxf(red[ch], red[ch + off]); __syncthreads(); }
    const float mx2 = red[0]; __syncthreads();
    const float e2 = __expf(m - mx2);
    red[ch] = e2; __syncthreads();
    for (int off = 32; off; off >>= 1) { if (ch < off) red[ch] += red[ch + off]; __syncthreads(); }
    const float sm2 = red[0]; __syncthreads();
    gmpW[b * 64 + ch] = e2 / sm2;
}

// ---------------------------------------------------------------------------
// Kernels 3a/3b: spatial softmax stats (max, sum-exp) for the 4 logit arrays.
// ---------------------------------------------------------------------------
__global__ __launch_bounds__(256) void k3a_lmax(
    const float* __restrict__ lA, const float* __restrict__ lB,
    const float* __restrict__ lC, const float* __restrict__ lD,
    float* __restrict__ lmax)
{
    const int T = blockIdx.x, b = blockIdx.y, tid = threadIdx.x;
    const float* l = (T == 0) ? lA : (T == 1) ? lB : (T == 2) ? lC : lD;
    float m = -3.0e38f;
    for (int idx = tid; idx < HWA; idx += 256) m = fmaxf(m, l[(size_t)b * HWA + idx]);
    __shared__ float red[256];
    red[tid] = m; __syncthreads();
    for (int off = 128; off; off >>= 1) { if (tid < off) red[tid] = fmaxf(red[tid], red[tid + off]); __syncthreads(); }
    if (tid == 0) lmax[b * 4 + T] = red[0];
}

__global__ __launch_bounds__(256) void k3b_lsum(
    const float* __restrict__ lA, const float* __restrict__ lB,
    const float* __restrict__ lC, const float* __restrict__ lD,
    const float* __restrict__ lmax, float* __restrict__ lsum)
{
    const int T = blockIdx.x, b = blockIdx.y, tid = threadIdx.x;
    const float* l = (T == 0) ? lA : (T == 1) ? lB : (T == 2) ? lC : lD;
    const float mx = lmax[b * 4 + T];
    float s = 0.f;
    for (int idx = tid; idx < HWA; idx += 256) s += __expf(l[(size_t)b * HWA + idx] - mx);
    __shared__ float red[256];
    red[tid] = s; __syncthreads();
    for (int off = 128; off; off >>= 1) { if (tid < off) red[tid] += red[tid + off]; __syncthreads(); }
    if (tid == 0) lsum[b * 4 + T] = red[0];
}

// ---------------------------------------------------------------------------
// Kernel 3c: fused pass over A1..D1: per-pixel gap/gmp projections and
// softmax-weighted per-channel sums (ordered partials for A2..D2).
// Block: (chunk of 256 pixels, b). Wave wv handles channels wv*8..wv*8+7.
// ---------------------------------------------------------------------------
__global__ __launch_bounds__(256) void k3c_proj(
    const float* __restrict__ A1, const float* __restrict__ B1,
    const float* __restrict__ C1, const float* __restrict__ D1,
    const float* __restrict__ lA, const float* __restrict__ lB,
    const float* __restrict__ lC, const float* __restrict__ lD,
    const float* __restrict__ gapW, const float* __restrict__ gmpW,
    const float* __restrict__ lmax, const float* __restrict__ lsum,
    float* __restrict__ t2part,
    float* __restrict__ vgap, float* __restrict__ vgmp,
    float* __restrict__ bgap, float* __restrict__ cgap, float* __restrict__ dgap)
{
    const int chunk = blockIdx.x, b = blockIdx.y, tid = threadIdx.x;
    const int lane = tid & 31, wv = tid >> 5;
    __shared__ float gA[64], gM[64];
    __shared__ float lmx[4], lis[4];
    __shared__ float lred[8][5][256];   // per-wave pixel partials (40 KB)

    if (tid < 64) { gA[tid] = gapW[b * 64 + tid]; gM[tid] = gmpW[b * 64 + tid]; }
    if (tid < 4)  { lmx[tid] = lmax[b * 4 + tid]; lis[tid] = 1.f / lsum[b * 4 + tid]; }
    __syncthreads();

    float t2[4][8];
    #pragma unroll
    for (int T = 0; T < 4; ++T)
        #pragma unroll
        for (int c8 = 0; c8 < 8; ++c8) t2[T][c8] = 0.f;
    float gp[8], gm[8], gbv[8], gcv[8], gdv[8];
    #pragma unroll
    for (int pp = 0; pp < 8; ++pp) { gp[pp] = gm[pp] = gbv[pp] = gcv[pp] = gdv[pp] = 0.f; }

    #pragma unroll
    for (int pp = 0; pp < 8; ++pp) {
        const int pix = chunk * 256 + pp * 32 + lane;
        const size_t lo = (size_t)b * HWA + pix;
        const float wsA = __expf(lA[lo] - lmx[0]) * lis[0];
        const float wsB = __expf(lB[lo] - lmx[1]) * lis[1];
        const float wsC = __expf(lC[lo] - lmx[2]) * lis[2];
        const float wsD = __expf(lD[lo] - lmx[3]) * lis[3];
        #pragma unroll
        for (int c8 = 0; c8 < 8; ++c8) {
            const int ch = wv * 8 + c8;
            const size_t oo = ((size_t)b * Cc + ch) * HWA + pix;
            const float a1 = A1[oo], b1 = B1[oo], c1 = C1[oo], d1 = D1[oo];
            t2[0][c8] += a1 * wsA; t2[1][c8] += b1 * wsB;
            t2[2][c8] += c1 * wsC; t2[3][c8] += d1 * wsD;
            const float ga = gA[ch], gmw = gM[ch];
            gp[pp]  += ga  * a1;  gm[pp]  += gmw * a1;
            gbv[pp] += ga  * b1;  gcv[pp] += ga  * c1;  gdv[pp] += ga * d1;
        }
    }
    #pragma unroll
    for (int pp = 0; pp < 8; ++pp) {
        lred[wv][0][pp * 32 + lane] = gp[pp];
        lred[wv][1][pp * 32 + lane] = gm[pp];
        lred[wv][2][pp * 32 + lane] = gbv[pp];
        lred[wv][3][pp * 32 + lane] = gcv[pp];
        lred[wv][4][pp * 32 + lane] = gdv[pp];
    }
    // wave-reduce the softmax-weighted channel sums -> ordered partials
    #pragma unroll
    for (int T = 0; T < 4; ++T)
        #pragma unroll
        for (int c8 = 0; c8 < 8; ++c8) {
            float v = t2[T][c8];
            #pragma unroll
            for (int off = 16; off; off >>= 1) v += __shfl_xor(v, off, 32);
            if (lane == 0)
                t2part[(((size_t)chunk * 8 + b) * 4 + T) * 64 + (wv * 8 + c8)] = v;
        }
    __syncthreads();
    float o0 = 0.f, o1 = 0.f, o2 = 0.f, o3 = 0.f, o4 = 0.f;
    #pragma unroll
    for (int w8 = 0; w8 < 8; ++w8) {
        o0 += lred[w8][0][tid]; o1 += lred[w8][1][tid]; o2 += lred[w8][2][tid];
        o3 += lred[w8][3][tid]; o4 += lred[w8][4][tid];
    }
    const size_t go = (size_t)b * HWA + chunk * 256 + tid;
    vgap[go] = o0; vgmp[go] = o1; bgap[go] = o2; cgap[go] = o3; dgap[go] = o4;
}

// ---------------------------------------------------------------------------
// Kernel 3d: reduce A2..D2 partials, 64x64 GEMV with m1..m4, silu -> fin.
// ---------------------------------------------------------------------------
__global__ __launch_bounds__(256) void k3d_fin(
    const float* __restrict__ t2part,
    const float* __restrict__ m1w, const float* __restrict__ m2w,
    const float* __restrict__ m3w, const float* __restrict__ m4w,
    float* __restrict__ fvals)
{
    const int b = blockIdx.x, tid = threadIdx.x;
    const int T = tid >> 6, ch = tid & 63;
    __shared__ float X2[4][64];
    float s = 0.f;
    for (int k = 0; k < 64; ++k)
        s += t2part[(((size_t)k * 8 + b) * 4 + T) * 64 + ch];
    X2[T][ch] = s;
    __syncthreads();
    const float* mw = (T == 0) ? m1w : (T == 1) ? m2w : (T == 2) ? m3w : m4w;
    float d = 0.f;
    for (int k = 0; k < 64; ++k) d += mw[ch * 64 + k] * X2[T][k];
    fvals[((size_t)b * 4 + T) * 64 + ch] = siluf(d);
}

// ---------------------------------------------------------------------------
// Kernel 4: per-pixel gates, closed-form iDWT, out = 1 - idwt + x.
// ---------------------------------------------------------------------------
__global__ __launch_bounds__(256) void k4_idwt_out(
    const float* __restrict__ x, const float* __restrict__ fvals,
    const float* __restrict__ vgap, const float* __restrict__ vgmp,
    const float* __restrict__ bgap, const float* __restrict__ cgap,
    const float* __restrict__ dgap,
    const float* __restrict__ c1w, const float* __restrict__ c1s, const float* __restrict__ c1t,
    float* __restrict__ out)
{
    const int i = blockIdx.x, b = blockIdx.y, tid = threadIdx.x;
    __shared__ float fsh[4][64];
    { const int T = tid >> 6, ch = tid & 63;
      fsh[T][ch] = fvals[((size_t)b * 4 + T) * 64 + ch]; }
    __syncthreads();

    const int col  = tid & 127;
    const int half = tid >> 7;          // two threads per pixel (ch halves)
    const size_t go = (size_t)b * HWA + (size_t)i * wAq + col;
    const float y  = (c1w[0] * vgap[go] + c1w[1] * vgmp[go]) * c1s[0] + c1t[0];
    const float aw = sigm(siluf(y));
    const float gb = sigm(bgap[go]);
    const float gc = sigm(cgap[go]);
    const float gd = sigm(dgap[go]);

    for (int cc = 0; cc < 32; ++cc) {
        const int ch = half * 32 + cc;
        const float FA = fsh[0][ch] * aw;
        const float FB = fsh[1][ch] * gb;
        const float FC = fsh[2][ch] * gc;
        const float FD = fsh[3][ch] * gd;
        const float o00 = 0.5f * (FA - FB - FC + FD);
        const float o01 = 0.5f * (FA + FB - FC - FD);
        const float o10 = 0.5f * (FA - FB + FC - FD);
        const float o11 = 0.5f * (FA + FB + FC + FD);
        const size_t xo = ((size_t)b * Cc + ch) * XHW + (size_t)(2 * i) * Wxx + 2 * col;
        const float2 xr0 = *(const float2*)(x + xo);
        const float2 xr1 = *(const float2*)(x + xo + Wxx);
        float2 r0; r0.x = 1.f - o00 + xr0.x; r0.y = 1.f - o01 + xr0.y;
        float2 r1; r1.x = 1.f - o10 + xr1.x; r1.y = 1.f - o11 + xr1.y;
        *(float2*)(out + xo)       = r0;
        *(float2*)(out + xo + Wxx) = r1;
    }
}

// ---------------------------------------------------------------------------
extern "C" void kernel_launch(void* const* d_in, const int* in_sizes, int n_in,
                              void* d_out, int out_size, void* d_ws, size_t ws_size,
                              hipStream_t stream) {
    (void)in_sizes; (void)n_in; (void)out_size; (void)ws_size;
    const float* x     = (const float*)d_in[0];
    const float* convw = (const float*)d_in[1];
    const float* convs = (const float*)d_in[2];
    const float* convt = (const float*)d_in[3];
    const float* c1w   = (const float*)d_in[4];
    const float* c1s   = (const float*)d_in[5];
    const float* c1t   = (const float*)d_in[6];
    const float* cAw   = (const float*)d_in[7];
    const float* cAs   = (const float*)d_in[8];
    const float* cAt   = (const float*)d_in[9];
    const float* cBw   = (const float*)d_in[10];
    const float* cBs   = (const float*)d_in[11];
    const float* cBt   = (const float*)d_in[12];
    const float* cCw   = (const float*)d_in[13];
    const float* cCs   = (const float*)d_in[14];
    const float* cCt   = (const float*)d_in[15];
    const float* cDw   = (const float*)d_in[16];
    const float* cDs   = (const float*)d_in[17];
    const float* cDt   = (const float*)d_in[18];
    const float* m1w   = (const float*)d_in[19];
    const float* m2w   = (const float*)d_in[20];
    const float* m3w   = (const float*)d_in[21];
    const float* m4w   = (const float*)d_in[22];
    float* out = (float*)d_out;

    // workspace carve-up (floats); total ~35.4M floats (~142 MB)
    float* W = (float*)d_ws;
    size_t o = 0;
    float* A1 = W + o;      o += (size_t)NB * Cc * HWA;
    float* B1 = W + o;      o += (size_t)NB * Cc * HWA;
    float* C1 = W + o;      o += (size_t)NB * Cc * HWA;
    float* D1 = W + o;      o += (size_t)NB * Cc * HWA;
    float* lA = W + o;      o += (size_t)NB * HWA;
    float* lB = W + o;      o += (size_t)NB * HWA;
    float* lC = W + o;      o += (size_t)NB * HWA;
    float* lD = W + o;      o += (size_t)NB * HWA;
    float* sumPart = W + o; o += (size_t)NB * 512 * 64;
    float* maxPart = W + o; o += (size_t)NB * 512 * 64;
    float* gapW = W + o;    o += (size_t)NB * 64;
    float* gmpW = W + o;    o += (size_t)NB * 64;
    float* lmax = W + o;    o += (size_t)NB * 4;
    float* lsum = W + o;    o += (size_t)NB * 4;
    float* t2part = W + o;  o += (size_t)64 * NB * 4 * 64;
    float* fvals = W + o;   o += (size_t)NB * 4 * 64;
    float* vgap = W + o;    o += (size_t)NB * HWA;
    float* vgmp = W + o;    o += (size_t)NB * HWA;
    float* bgap = W + o;    o += (size_t)NB * HWA;
    float* cgap = W + o;    o += (size_t)NB * HWA;
    float* dgap = W + o;    o += (size_t)NB * HWA;

    k1_conv_dwt<<<dim3(4, 128, 8), 256, 0, stream>>>(
        x, convw, convs, convt,
        cAw, cAs, cAt, cBw, cBs, cBt, cCw, cCs, cCt, cDw, cDs, cDt,
        A1, B1, C1, D1, lA, lB, lC, lD, sumPart, maxPart);
    k2_gap<<<dim3(8), 64, 0, stream>>>(sumPart, maxPart, gapW, gmpW);
    k3a_lmax<<<dim3(4, 8), 256, 0, stream>>>(lA, lB, lC, lD, lmax);
    k3b_lsum<<<dim3(4, 8), 256, 0, stream>>>(lA, lB, lC, lD, lmax, lsum);
    k3c_proj<<<dim3(64, 8), 256, 0, stream>>>(
        A1, B1, C1, D1, lA, lB, lC, lD, gapW, gmpW, lmax, lsum,
        t2part, vgap, vgmp, bgap, cgap, dgap);
    k3d_fin<<<dim3(8), 256, 0, stream>>>(t2part, m1w, m2w, m3w, m4w, fvals);
    k4_idwt_out<<<dim3(128, 8), 256, 0, stream>>>(
        x, fvals, vgap, vgmp, bgap, cgap, dgap, c1w, c1s, c1t, out);
}